// AttnCLRLoss_91001767067770
// MI455X (gfx1250) — compile-verified
//
#include <hip/hip_runtime.h>
#include <cstddef>

typedef float v2f __attribute__((ext_vector_type(2)));
typedef float v8f __attribute__((ext_vector_type(8)));
typedef int   v4i __attribute__((ext_vector_type(4)));

#define N_ROWS 4096
#define BHALF  2048
#define D_DIM  128
#define TINV   (1.0f / 0.07f)
#define JSPLIT 8
#define JT_PER (256 / JSPLIT)   // 32 j-tiles per chunk

#define GAS __attribute__((address_space(1)))   // global
#define LAS __attribute__((address_space(3)))   // LDS

#if __has_builtin(__builtin_amdgcn_global_load_async_to_lds_b128) && \
    __has_builtin(__builtin_amdgcn_s_wait_asynccnt)
#define USE_ASYNC_LDS 1
#else
#define USE_ASYNC_LDS 0
#endif

// ---------------- Kernel A: L2 row-normalize features ----------------
__global__ void k_normalize(const float* __restrict__ feat, float* __restrict__ fn) {
  const int row = blockIdx.x;      // 4096 rows
  const int t   = threadIdx.x;     // 128 threads = D
  float v = feat[(size_t)row * D_DIM + t];
  float s = v * v;
  #pragma unroll
  for (int m = 16; m >= 1; m >>= 1) s += __shfl_xor(s, m, 32);
  __shared__ float red[4];
  if ((t & 31) == 0) red[t >> 5] = s;
  __syncthreads();
  float tot = red[0] + red[1] + red[2] + red[3];
  fn[(size_t)row * D_DIM + t] = v / sqrtf(tot);
}

// ---------------- block reductions for 256-thread (8-wave) blocks ----------------
__device__ __forceinline__ float blk_sum256(float v, float* red) {
  #pragma unroll
  for (int m = 16; m >= 1; m >>= 1) v += __shfl_xor(v, m, 32);
  const int t = threadIdx.x;
  if ((t & 31) == 0) red[t >> 5] = v;
  __syncthreads();
  float s = 0.f;
  #pragma unroll
  for (int w = 0; w < 8; ++w) s += red[w];
  __syncthreads();
  return s;
}

__device__ __forceinline__ float blk_max256(float v, float* red) {
  #pragma unroll
  for (int m = 16; m >= 1; m >>= 1) v = fmaxf(v, __shfl_xor(v, m, 32));
  const int t = threadIdx.x;
  if ((t & 31) == 0) red[t >> 5] = v;
  __syncthreads();
  float s = red[0];
  #pragma unroll
  for (int w = 1; w < 8; ++w) s = fmaxf(s, red[w]);
  __syncthreads();
  return s;
}

// ---------------- Kernel B: row-wise sparsemax of (attn * negmask / T) ----------------
// Row held entirely in registers (16 values/thread); reductions via shuffles.
// attn is single-use -> non-temporal loads keep L2 free for masked_scores reuse.
__global__ void k_sparsemax(const float* __restrict__ attn, float* __restrict__ ms) {
  const int i = blockIdx.x;        // row
  const int t = threadIdx.x;       // 256 threads
  __shared__ float red[8];
  const int ib = i & (BHALF - 1);
  const float* __restrict__ rowp = attn + (size_t)i * N_ROWS;

  __builtin_prefetch(&rowp[t], 0, 0);

  float zr[16];
  float lmax = -3.4e38f;
  #pragma unroll
  for (int k = 0; k < 16; ++k) {
    const int j = t + 256 * k;
    float a  = __builtin_nontemporal_load(&rowp[j]);
    float zz = ((j & (BHALF - 1)) == ib) ? 0.0f : a * TINV;
    zr[k] = zz;
    lmax = fmaxf(lmax, zz);
  }
  float zmax = blk_max256(lmax, red);

  // Binary search tau in [zmax-1, zmax]; S(tau)=sum(max(z-tau,0)) is monotone decreasing.
  float lo = zmax - 1.0f, hi = zmax;
  for (int it = 0; it < 24; ++it) {
    float tau = 0.5f * (lo + hi);
    float p = 0.f;
    #pragma unroll
    for (int k = 0; k < 16; ++k) p += fmaxf(zr[k] - tau, 0.0f);
    float s = blk_sum256(p, red);
    if (s > 1.0f) lo = tau; else hi = tau;
  }
  // Exact tau from the resolved support set.
  const float thr = 0.5f * (lo + hi);
  float c = 0.f, su = 0.f;
  #pragma unroll
  for (int k = 0; k < 16; ++k) {
    if (zr[k] > thr) { c += 1.0f; su += zr[k]; }
  }
  float cnt = blk_sum256(c, red);
  float sum = blk_sum256(su, red);
  const float tau = (sum - 1.0f) / cnt;
  float* __restrict__ orow = ms + (size_t)i * N_ROWS;
  #pragma unroll
  for (int k = 0; k < 16; ++k) orow[t + 256 * k] = fmaxf(zr[k] - tau, 0.0f);
}

// ---------------- Kernel C: fused Gram GEMM (f32 WMMA) + weighted exp partial sums ----
// Block = 8 waves; wave w owns itile = blockIdx.x*8+w; blockIdx.y = j-chunk (32 j-tiles).
// B tile (16x128, 8 KB) staged in LDS, double-buffered, shared by all 8 waves.
// Staging uses GLOBAL_LOAD_ASYNC_TO_LDS (ASYNCcnt) when available: L2 -> LDS with no
// VGPR round-trip; overlap expressed via s_wait_asynccnt + split barriers.
__global__ void __launch_bounds__(256)
k_gemm_loss(const float* __restrict__ fn, const float* __restrict__ ms,
            float* __restrict__ part, float* __restrict__ spg) {
  const int t      = threadIdx.x;
  const int wave   = t >> 5;
  const int lane   = t & 31;
  const int half   = lane >> 4;
  const int lp     = lane & 15;
  const int itile  = blockIdx.x * 8 + wave;
  const int jchunk = blockIdx.y;
  const int i16    = itile * 16;
  const int jt0    = jchunk * JT_PER, jt1 = jt0 + JT_PER;

  __shared__ float tB[2][16 * D_DIM];   // 2 x 8 KB double buffer

  // Resident A fragments: a[ks][v] = fn[i16+lp][4*ks + v + 2*half]
  v2f a[32];
  const float* arow = fn + (size_t)(i16 + lp) * D_DIM + 2 * half;
  #pragma unroll
  for (int ks = 0; ks < 32; ++ks) a[ks] = *(const v2f*)(arow + 4 * ks);

  float rowpart[8];
  #pragma unroll
  for (int r = 0; r < 8; ++r) rowpart[r] = 0.f;

  const int pjtile = (itile + 128) & 255;

#if USE_ASYNC_LDS
  // ---- async double-buffer pipeline ----
  {
    GAS v4i* g = (GAS v4i*)(fn + (size_t)jt0 * 16 * D_DIM);
    LAS v4i* l = (LAS v4i*)tB[0];
    __builtin_amdgcn_global_load_async_to_lds_b128(g + t,       l + t,       0, 0);
    __builtin_amdgcn_global_load_async_to_lds_b128(g + t + 256, l + t + 256, 0, 0);
  }
  for (int jt = jt0; jt < jt1; ++jt) {
    const int  cur  = (jt - jt0) & 1;
    const bool more = (jt + 1 < jt1);
    __builtin_amdgcn_s_wait_asynccnt(0);   // my writes to tB[cur] landed
    __syncthreads();                       // everyone's landed; other buffer free
    if (more) {
      GAS v4i* g = (GAS v4i*)(fn + (size_t)(jt + 1) * 16 * D_DIM);
      LAS v4i* l = (LAS v4i*)tB[cur ^ 1];
      __builtin_amdgcn_global_load_async_to_lds_b128(g + t,       l + t,       0, 0);
      __builtin_amdgcn_global_load_async_to_lds_b128(g + t + 256, l + t + 256, 0, 0);
    }
#else
  // ---- register-staged double-buffer pipeline (fallback) ----
  {
    const float4* src = (const float4*)(fn + (size_t)jt0 * 16 * D_DIM);
    float4* dst = (float4*)tB[0];
    dst[t]       = src[t];
    dst[t + 256] = src[t + 256];
  }
  float4 nx0, nx1;
  for (int jt = jt0; jt < jt1; ++jt) {
    const int  cur  = (jt - jt0) & 1;
    const bool more = (jt + 1 < jt1);
    if (more) {
      const float4* src = (const float4*)(fn + (size_t)(jt + 1) * 16 * D_DIM);
      nx0 = src[t];
      nx1 = src[t + 256];
    }
    __syncthreads();
#endif

    // Batch-load B fragments from LDS, then run the WMMA chain.
    v2f b[32];
    const float* bl = tB[cur] + lp * D_DIM + 2 * half;
    #pragma unroll
    for (int ks = 0; ks < 32; ++ks) b[ks] = *(const v2f*)(bl + 4 * ks);

    v8f acc = {0.f, 0.f, 0.f, 0.f, 0.f, 0.f, 0.f, 0.f};
    #pragma unroll
    for (int ks = 0; ks < 32; ++ks)
      acc = __builtin_amdgcn_wmma_f32_16x16x4_f32(
          false, a[ks], false, b[ks], (short)0, acc, false, false);

    // D layout: lane, vgpr r -> M = r + 8*half, N = lp.
    const int j = jt * 16 + lp;
    if (jt == pjtile) {   // partner column n == m within this tile; unique writer grid-wide
      #pragma unroll
      for (int r = 0; r < 8; ++r) {
        const int m = r + 8 * half;
        if (lp == m) spg[i16 + m] = acc[r];
      }
    }
    #pragma unroll
    for (int r = 0; r < 8; ++r) {
      const int gi = i16 + r + 8 * half;
      const float w = ((gi == j) ? 0.0f : 1.0f) - ms[(size_t)gi * N_ROWS + j];
      rowpart[r] += __expf((acc[r] - 1.0f) * TINV) * w;   // rowmax == 1/T analytically
    }

#if USE_ASYNC_LDS
  }
#else
    if (more) {
      float4* dst = (float4*)tB[cur ^ 1];
      dst[t]       = nx0;
      dst[t + 256] = nx1;
    }
  }
#endif

  // Reduce each rowpart[r] across the 16 lanes of its half-wave, emit partials.
  #pragma unroll
  for (int r = 0; r < 8; ++r) {
    float v = rowpart[r];
    #pragma unroll
    for (int m = 8; m >= 1; m >>= 1) v += __shfl_xor(v, m, 16);
    if (lp == 0)
      part[((size_t)itile * JSPLIT + jchunk) * 16 + r + 8 * half] = v;
  }
}

// ---------------- Kernel D: combine partials, per-row log-prob, final mean ----------
__global__ void k_combine(const float* __restrict__ part, const float* __restrict__ spg,
                          float* __restrict__ out) {
  const int t = threadIdx.x;    // 1024 threads = 32 waves
  float s = 0.f;
  for (int row = t; row < N_ROWS; row += 1024) {
    const int it = row >> 4, m = row & 15;
    float rs = 0.f;
    #pragma unroll
    for (int c = 0; c < JSPLIT; ++c) rs += part[((size_t)it * JSPLIT + c) * 16 + m];
    s += (spg[row] - 1.0f) * TINV - __logf(rs);
  }
  #pragma unroll
  for (int m = 16; m >= 1; m >>= 1) s += __shfl_xor(s, m, 32);
  __shared__ float red[32];
  if ((t & 31) == 0) red[t >> 5] = s;
  __syncthreads();
  if (t < 32) {
    float v = red[t];
    #pragma unroll
    for (int m = 16; m >= 1; m >>= 1) v += __shfl_xor(v, m, 32);
    if (t == 0) out[0] = -v / (float)N_ROWS;
  }
}

extern "C" void kernel_launch(void* const* d_in, const int* in_sizes, int n_in,
                              void* d_out, int out_size, void* d_ws, size_t ws_size,
                              hipStream_t stream) {
  const float* feat = (const float*)d_in[0];   // (2048, 2, 128) f32, flat == (4096, 128)
  const float* attn = (const float*)d_in[1];   // (4096, 4096) f32
  float* out = (float*)d_out;                  // [0] = loss, [1..] = masked_scores
  float* msc = out + 1;

  float* fn   = (float*)d_ws;                        // 4096*128 floats  (2 MB)
  float* part = fn + (size_t)N_ROWS * D_DIM;         // 256*8*16 floats  (128 KB)
  float* spg  = part + (size_t)256 * JSPLIT * 16;    // 4096 floats      (16 KB)

  k_normalize<<<N_ROWS, D_DIM, 0, stream>>>(feat, fn);
  k_sparsemax<<<N_ROWS, 256, 0, stream>>>(attn, msc);
  k_gemm_loss<<<dim3(32, JSPLIT), 256, 0, stream>>>(fn, msc, part, spg);
  k_combine<<<1, 1024, 0, stream>>>(part, spg, out);
}